// QuantizationLayer_33277406609706
// MI455X (gfx1250) — compile-verified
//
#include <hip/hip_runtime.h>

typedef float v2f __attribute__((ext_vector_type(2)));
typedef float v8f __attribute__((ext_vector_type(8)));

#define B_    16
#define N_    1048576
#define S_    32
#define SEG_  32768
#define DIM_  256
#define BINS_ 8192      // S_*DIM_
#define CONT_ 65536     // DIM_*DIM_
#define VERW_ 512       // 16384 bits / 32

// ---------------------------------------------------------------------------
// K1: per (batch, segment) 256-bin histogram for x and y.
// Block (s, b) exclusively owns hist[b][s*256 .. s*256+255] for each axis,
// so LDS atomics + plain global stores (no global atomics, no init needed).
// ---------------------------------------------------------------------------
__global__ __launch_bounds__(256) void hist_kernel(const int* __restrict__ xs,
                                                   const int* __restrict__ ys,
                                                   unsigned* __restrict__ hx,
                                                   unsigned* __restrict__ hy) {
  __shared__ unsigned lx[DIM_];
  __shared__ unsigned ly[DIM_];
  const int s = blockIdx.x, b = blockIdx.y, tid = threadIdx.x;
  lx[tid] = 0u; ly[tid] = 0u;
  __syncthreads();

  const size_t base = (size_t)b * N_ + (size_t)s * SEG_;
  const int4* x4 = (const int4*)(xs + base);
  const int4* y4 = (const int4*)(ys + base);
  for (int i = tid; i < SEG_ / 4; i += 256) {
    if (i + 64 < SEG_ / 4) {
      __builtin_prefetch(&x4[i + 64], 0, 0);   // global_prefetch_b8
      __builtin_prefetch(&y4[i + 64], 0, 0);
    }
    int4 vx = x4[i];
    int4 vy = y4[i];
    atomicAdd(&lx[vx.x], 1u); atomicAdd(&lx[vx.y], 1u);
    atomicAdd(&lx[vx.z], 1u); atomicAdd(&lx[vx.w], 1u);
    atomicAdd(&ly[vy.x], 1u); atomicAdd(&ly[vy.y], 1u);
    atomicAdd(&ly[vy.z], 1u); atomicAdd(&ly[vy.w], 1u);
  }
  __syncthreads();
  const size_t o = ((size_t)b * S_ + s) * DIM_ + tid;
  hx[o] = lx[tid];
  hy[o] = ly[tid];
}

// ---------------------------------------------------------------------------
// K2: per (batch, axis): stats -> clip -> 3x3 blur -> WMMA centroid -> shift.
// shift[b,s] = round(m[b,s] - 128)   (the "start" term cancels algebraically)
// Centroid m = blur(32x256) x weights(256) done with v_wmma_f32_16x16x4_f32:
// B-matrix broadcasts the weight pair across all 16 N-columns, so every
// column of D holds the same row-dot; we read column 0.
// ---------------------------------------------------------------------------
__global__ __launch_bounds__(256) void shift_kernel(const unsigned* __restrict__ hx,
                                                    const unsigned* __restrict__ hy,
                                                    const float* __restrict__ bk,
                                                    int* __restrict__ shifts) {
  extern __shared__ float sm2[];
  float* img = sm2;              // 8192
  float* blr = img + BINS_;      // 8192
  float* red = blr + BINS_;      // 256
  float* mv  = red + 256;        // 32

  const int tid = threadIdx.x;
  const int b = blockIdx.x, axis = blockIdx.y;
  const unsigned* h = (axis == 0 ? hx : hy) + (size_t)b * BINS_;

  // mean
  float part = 0.f;
  for (int i = tid; i < BINS_; i += 256) {
    float v = (float)h[i];
    img[i] = v;
    part += v;
  }
  red[tid] = part;
  __syncthreads();
  for (int st = 128; st > 0; st >>= 1) {
    if (tid < st) red[tid] += red[tid + st];
    __syncthreads();
  }
  const float mean = red[0] * (1.0f / (float)BINS_);
  __syncthreads();

  // std (ddof=1), two-pass
  part = 0.f;
  for (int i = tid; i < BINS_; i += 256) {
    float d = img[i] - mean;
    part += d * d;
  }
  red[tid] = part;
  __syncthreads();
  for (int st = 128; st > 0; st >>= 1) {
    if (tid < st) red[tid] += red[tid + st];
    __syncthreads();
  }
  const float stdv = sqrtf(red[0] / (float)(BINS_ - 1));
  const float hi = mean + 3.0f * stdv;
  __syncthreads();

  // clip
  for (int i = tid; i < BINS_; i += 256) {
    float v = img[i];
    v = v < 0.f ? 0.f : v;
    v = v > hi ? hi : v;
    img[i] = v;
  }
  __syncthreads();

  // 3x3 blur with zero padding (correlation; kernel is symmetric anyway)
  float kk[9];
#pragma unroll
  for (int j = 0; j < 9; ++j) kk[j] = bk[j];
  for (int idx = tid; idx < BINS_; idx += 256) {
    const int s = idx >> 8, w = idx & 255;
    float acc = 0.f;
#pragma unroll
    for (int dy = 0; dy < 3; ++dy) {
      const int ss = s + dy - 1;
      if (ss < 0 || ss >= S_) continue;
#pragma unroll
      for (int dx = 0; dx < 3; ++dx) {
        const int ww = w + dx - 1;
        if (ww < 0 || ww >= DIM_) continue;
        acc += kk[dy * 3 + dx] * img[ss * DIM_ + ww];
      }
    }
    blr[idx] = acc;
  }
  __syncthreads();

  // WMMA centroid on wave 0 (EXEC all ones within the wave).
  // A 16x4 f32: lane<16 -> {A[M=lane][0],A[lane][1]}, lane>=16 -> {A[lane-16][2],[3]}
  // B  4x16 f32 (mirror), broadcast weight across N columns.
  // D 16x16 f32: vgpr r: lanes 0-15 -> M=r, lanes 16-31 -> M=r+8.
  if (tid < 32) {
    const int lane = tid;
    const int half = lane >> 4;        // 0 or 1
    const int mrow = lane & 15;
    for (int g = 0; g < 2; ++g) {
      v8f c = {0.f, 0.f, 0.f, 0.f, 0.f, 0.f, 0.f, 0.f};
      const int row = g * 16 + mrow;
      for (int ks = 0; ks < 64; ++ks) {
        const int wb = ks * 4 + half * 2;
        v2f a;
        a.x = blr[row * DIM_ + wb];
        a.y = blr[row * DIM_ + wb + 1];
        v2f bb;
        bb.x = (float)wb;
        bb.y = (float)(wb + 1);
        c = __builtin_amdgcn_wmma_f32_16x16x4_f32(false, a, false, bb,
                                                  (short)0, c, false, false);
      }
      if (lane == 0) {
#pragma unroll
        for (int r = 0; r < 8; ++r) mv[g * 16 + r] = c[r];
      } else if (lane == 16) {
#pragma unroll
        for (int r = 0; r < 8; ++r) mv[g * 16 + 8 + r] = c[r];
      }
    }
  }
  __syncthreads();

  if (tid < 32) {
    const float m = mv[tid] * (1.0f / (float)SEG_);
    shifts[((size_t)axis * B_ + b) * S_ + tid] = (int)rintf(m - (float)(DIM_ / 2));
  }
}

// ---------------------------------------------------------------------------
// K3: per-batch sequential accumulation, entirely in LDS (CDNA5: 320KB/WGP).
// container: 65536 u32 (256KB), verifier: 2x512 u32 bitsets (4KB).
// Per segment: tentative ds_or into tmp bitset counting first-setters
// (= distinct newly-set bits = xor_cnt), uniform ratio test, then commit
// container ds_adds from register-held indices + buffer swap, or break.
// ---------------------------------------------------------------------------
__global__ __launch_bounds__(1024) void accum_kernel(const int* __restrict__ xs,
                                                     const int* __restrict__ ys,
                                                     const int* __restrict__ shifts,
                                                     float* __restrict__ out) {
  extern __shared__ unsigned sm3[];
  unsigned* cont = sm3;            // 65536
  unsigned* verA = cont + CONT_;   // 512
  unsigned* verB = verA + VERW_;   // 512

  __shared__ int sxv[S_];
  __shared__ int syv[S_];
  __shared__ unsigned cnt_new;
  __shared__ unsigned prev_cnt;

  const int b = blockIdx.x;
  const int tid = threadIdx.x;

  if (tid < S_) {
    sxv[tid] = shifts[(0 * B_ + b) * S_ + tid];
    syv[tid] = shifts[(1 * B_ + b) * S_ + tid];
  }
  for (int i = tid; i < CONT_; i += 1024) cont[i] = 0u;
  for (int i = tid; i < VERW_; i += 1024) verA[i] = 0u;
  if (tid == 0) cnt_new = 0u;
  __syncthreads();

  const size_t base0 = (size_t)b * N_;

  // ---- segment START=2: seed container + verifier ----
  {
    const int sx = sxv[2], sy = syv[2];
    const size_t segbase = base0 + 2 * (size_t)SEG_;
#pragma unroll
    for (int j = 0; j < 32; ++j) {
      const size_t e = segbase + (size_t)j * 1024 + tid;
      int x = xs[e] - sx;  x = x < 0 ? 0 : (x > 255 ? 255 : x);
      int y = ys[e] - sy;  y = y < 0 ? 0 : (y > 255 ? 255 : y);
      const int ic = x + (y << 8);
      atomicAdd(&cont[ic], 1u);
      const int iv = (x >> 1) + ((y >> 1) << 7);
      atomicOr(&verA[iv >> 5], 1u << (iv & 31));
    }
  }
  __syncthreads();

  // popcount(verifier)
  {
    unsigned p = 0;
    for (int i = tid; i < VERW_; i += 1024) p += (unsigned)__popc((int)verA[i]);
    atomicAdd(&cnt_new, p);
  }
  __syncthreads();
  if (tid == 0) prev_cnt = cnt_new;
  __syncthreads();

  unsigned* cur = verA;
  unsigned* tmp = verB;

  // ---- segments 3..31: tentative update, uniform test, commit or break ----
  for (int s = 3; s < S_; ++s) {
    for (int i = tid; i < VERW_; i += 1024) tmp[i] = cur[i];
    if (tid == 0) cnt_new = 0u;
    __syncthreads();

    const int sx = sxv[s], sy = syv[s];
    const size_t segbase = base0 + (size_t)s * SEG_;
    int icl[32];
    unsigned localnew = 0u;
#pragma unroll
    for (int j = 0; j < 32; ++j) {
      const size_t e = segbase + (size_t)j * 1024 + tid;
      int x = xs[e] - sx;  x = x < 0 ? 0 : (x > 255 ? 255 : x);
      int y = ys[e] - sy;  y = y < 0 ? 0 : (y > 255 ? 255 : y);
      const int ic = x + (y << 8);
      icl[j] = ic;
      const int iv = (x >> 1) + ((y >> 1) << 7);
      const unsigned bit = 1u << (iv & 31);
      const unsigned old = atomicOr(&tmp[iv >> 5], bit);
      localnew += ((old & bit) == 0u) ? 1u : 0u;
    }
    atomicAdd(&cnt_new, localnew);
    __syncthreads();

    const unsigned xorc = cnt_new;               // distinct newly-set bits
    const unsigned newc = prev_cnt + xorc;       // popcount after update
    const float ratio = (float)xorc / (float)newc;
    if (ratio < 0.01f) break;                    // uniform across block

#pragma unroll
    for (int j = 0; j < 32; ++j) atomicAdd(&cont[icl[j]], 1u);

    unsigned* t = cur; cur = tmp; tmp = t;       // commit verifier
    if (tid == 0) prev_cnt = newc;
    __syncthreads();
  }
  __syncthreads();

  float* ob = out + (size_t)b * CONT_;
  for (int i = tid; i < CONT_; i += 1024) ob[i] = (float)cont[i];
}

// ---------------------------------------------------------------------------
extern "C" void kernel_launch(void* const* d_in, const int* in_sizes, int n_in,
                              void* d_out, int out_size, void* d_ws, size_t ws_size,
                              hipStream_t stream) {
  (void)in_sizes; (void)n_in; (void)out_size; (void)ws_size;
  const int*   xs = (const int*)d_in[0];
  const int*   ys = (const int*)d_in[1];
  const float* bk = (const float*)d_in[2];
  float* out = (float*)d_out;

  unsigned* hx = (unsigned*)d_ws;                 // 16*8192 u32 = 512KB
  unsigned* hy = hx + (size_t)B_ * BINS_;         // 512KB
  int* shifts  = (int*)(hy + (size_t)B_ * BINS_); // 2*16*32 ints = 4KB

  hist_kernel<<<dim3(S_, B_), 256, 0, stream>>>(xs, ys, hx, hy);

  const size_t sm2 = (size_t)(BINS_ + BINS_ + 256 + 32) * sizeof(float); // 66,688 B
  shift_kernel<<<dim3(B_, 2), 256, sm2, stream>>>(hx, hy, bk, shifts);

  const size_t sm3 = (size_t)(CONT_ + 2 * VERW_) * sizeof(unsigned);    // 266,240 B
  accum_kernel<<<dim3(B_), 1024, sm3, stream>>>(xs, ys, shifts, out);
}